// MeruSegformer_76733885710934
// MI455X (gfx1250) — compile-verified
//
#include <hip/hip_runtime.h>
#include <math.h>

// ---------------------------------------------------------------------------
// MERU segmentation loss for gfx1250 (MI455X), wave32.
// GEMM core: V_WMMA_F32_16X16X4_F32 (full fp32; kernel is HBM/VALU bound,
// not matrix-bound, so fp32 WMMA costs nothing vs fp16).
// Epilogue is fully branchless (v_cndmask) — one v_log + one v_exp per
// (pixel,class) element.
// ---------------------------------------------------------------------------

typedef __attribute__((ext_vector_type(2))) float v2f;
typedef __attribute__((ext_vector_type(8))) float v8f;

#define F_EPS        1e-8f
#define SINH_MAXF    11.090354888959125f   /* asinh(2^15) */
#define LOGIT_SCALE  (1.0f / 0.07f)
#define IMG_ALPHA    0.25f
#define TXT_ALPHA    (1.0f / 0.6f)
#define ENT_W        0.2f
#define C_CLS        151
#define C_PAD        160
#define D_DIM        64
#define ROW          68                    /* padded LDS row stride (dwords) */
#define PIX_PER_WG   128

// ---------------------------------------------------------------------------
// Kernel 1: per-class precompute. txt = exp_map0(protos/0.6), y_time, |txt|,
// half-aperture. Written to workspace; tiny (151x64).
// ---------------------------------------------------------------------------
__global__ void meru_protos(const float* __restrict__ tp, float* __restrict__ ws) {
  int c = threadIdx.x;
  if (c >= C_PAD) return;
  float* txt = ws;
  float* yt  = ws + C_PAD * D_DIM;
  float* pn  = yt + C_PAD;
  float* ap  = pn + C_PAD;
  if (c < C_CLS) {
    float n2 = 0.0f;
    for (int d = 0; d < D_DIM; ++d) {
      float y = tp[c * D_DIM + d] * TXT_ALPHA;
      n2 += y * y;
    }
    float rc = sqrtf(n2);
    float s  = fminf(fmaxf(rc, F_EPS), SINH_MAXF);
    float sc = sinhf(s) / fmaxf(rc, F_EPS);
    float tn2 = 0.0f;
    for (int d = 0; d < D_DIM; ++d) {
      float t = tp[c * D_DIM + d] * TXT_ALPHA * sc;
      txt[c * D_DIM + d] = t;
      tn2 += t * t;
    }
    float tn = sqrtf(tn2);
    yt[c] = sqrtf(1.0f + tn2);
    pn[c] = tn;
    float ai = fminf(fmaxf(0.2f / (tn + F_EPS), -1.0f + F_EPS), 1.0f - F_EPS);
    ap[c] = asinf(ai);
  } else {
    for (int d = 0; d < D_DIM; ++d) txt[c * D_DIM + d] = 0.0f;
    yt[c] = 1.0f; pn[c] = 1.0f; ap[c] = 0.0f;
  }
}

// ---------------------------------------------------------------------------
// Kernel 2: main fused kernel. One WG (256 thr = 8 waves) per 128 pixels.
// Wave w owns pixels [16w,16w+16): exp-map into register A-fragments,
// 10 N-tiles x 16 K-steps of v_wmma_f32_16x16x4_f32 against prototypes in
// LDS (padded rows -> conflict-free ds_load_2addr_b64), then a branchless
// online softmax + label gather folded into the accumulator layout,
// xor-shuffle reductions, per-WG partials to workspace.
// ---------------------------------------------------------------------------
__global__ __launch_bounds__(256) void meru_main(
    const float* __restrict__ feats, const int* __restrict__ labels,
    const unsigned char* __restrict__ mask, const float* __restrict__ ws,
    float* __restrict__ partials) {
  __shared__ __align__(16) float b_lds[C_PAD * ROW];
  __shared__ float yt_l[C_PAD], pn_l[C_PAD], ap_l[C_PAD];
  __shared__ float xt_l[PIX_PER_WG];
  __shared__ int   lab_l[PIX_PER_WG];
  __shared__ float vld_l[PIX_PER_WG];
  __shared__ float red[256 * 3];

  const int tid = threadIdx.x;
  const float* txt = ws;
  const float* yt  = ws + C_PAD * D_DIM;
  const float* pn  = yt + C_PAD;
  const float* ap  = pn + C_PAD;

  for (int i = tid; i < C_PAD * D_DIM; i += 256)
    b_lds[(i >> 6) * ROW + (i & 63)] = txt[i];
  for (int i = tid; i < C_PAD; i += 256) {
    yt_l[i] = yt[i]; pn_l[i] = pn[i]; ap_l[i] = ap[i];
  }

  const int wave = tid >> 5;
  const int lane = tid & 31;
  const int lm   = lane & 15;        // A: pixel row / B,C,D: column index
  const int hi   = lane >> 4;        // lane-half selects K pair / M+8
  const long long p = (long long)blockIdx.x * PIX_PER_WG + wave * 16 + lm;

  // ---- phase 1: exp_map0(feats * 0.25) straight into A fragments ----------
  v2f a2[16];
  float n2p = 0.0f;
  const float* fp = feats + p * D_DIM + 2 * hi;
#pragma unroll
  for (int k = 0; k < 16; ++k) {
    v2f x = *(const v2f*)(fp + 4 * k);
    x *= IMG_ALPHA;
    a2[k] = x;
    n2p += x.x * x.x + x.y * x.y;
  }
  float n2 = n2p + __shfl_xor(n2p, 16);      // combine lane l with l+16
  float rc = sqrtf(n2);
  float s  = fminf(fmaxf(rc, F_EPS), SINH_MAXF);
  float sc = sinhf(s) / fmaxf(rc, F_EPS);
#pragma unroll
  for (int k = 0; k < 16; ++k) a2[k] *= sc;
  float xt = sqrtf(1.0f + sc * sc * n2);     // x_time of this pixel
  if (hi == 0) {
    int pw = wave * 16 + lm;
    xt_l[pw]  = xt;
    lab_l[pw] = labels[p];
    vld_l[pw] = mask[p] ? 0.0f : 1.0f;
  }
  __syncthreads();

  float xtv[8];
  int   labv[8];
#pragma unroll
  for (int v = 0; v < 8; ++v) {
    int pm = wave * 16 + v + 8 * hi;         // accumulator row M for this lane
    xtv[v]  = xt_l[pm];
    labv[v] = lab_l[pm];
  }

  float mrun[8], srun[8], dlab[8], cxlab[8];
#pragma unroll
  for (int v = 0; v < 8; ++v) {
    mrun[v] = -INFINITY; srun[v] = 0.0f; dlab[v] = 0.0f; cxlab[v] = 0.0f;
  }

  // ---- phase 2: 10 class tiles x 16 K-steps of WMMA -----------------------
  for (int t = 0; t < 10; ++t) {
    const int   c   = 16 * t + lm;
    const float ytc = yt_l[c];
    const bool  cv  = (c < C_CLS);
    v8f acc = {0.0f, 0.0f, 0.0f, 0.0f, 0.0f, 0.0f, 0.0f, 0.0f};
    const float* bp = &b_lds[c * ROW + 2 * hi];
#pragma unroll
    for (int k = 0; k < 16; ++k) {
      v2f b = *(const v2f*)(bp + 4 * k);
      acc = __builtin_amdgcn_wmma_f32_16x16x4_f32(
          false, a2[k], false, b, (short)0, acc, false, false);
    }
    // Branchless epilogue: one v_log + one v_exp per (pixel,class).
#pragma unroll
    for (int v = 0; v < 8; ++v) {
      float inner = acc[v];                        // <img_m, txt_c>
      float cx    = fmaf(xtv[v], ytc, -inner);     // -Lorentz inner (>= 1)
      float cxc   = fmaxf(cx, 1.0f + F_EPS);
      float w     = cxc + sqrtf(fmaf(cxc, cxc, -1.0f));
      float dist  = __logf(w);                     // arccosh
      float z     = cv ? (-LOGIT_SCALE * dist) : -INFINITY;
      // online softmax, single exp, no branches:
      float mo = fmaxf(mrun[v], z);
      float dd = fminf(mrun[v], z) - mo;           // <= 0 (or -INF)
      float e  = __expf(dd);
      srun[v]  = (z > mrun[v]) ? fmaf(srun[v], e, 1.0f) : (srun[v] + e);
      mrun[v]  = mo;
      // label gather as select + fma:
      float sel = (c == labv[v]) ? 1.0f : 0.0f;
      dlab[v]  = fmaf(sel, dist, dlab[v]);
      cxlab[v] = fmaf(sel, cx, cxlab[v]);
    }
  }

  // ---- phase 3: reduce across the 16 class lanes (xor <= 8 stays in half) -
#pragma unroll
  for (int v = 0; v < 8; ++v) {
    float m = mrun[v], sg = srun[v], dl = dlab[v], cl = cxlab[v];
#pragma unroll
    for (int off = 8; off >= 1; off >>= 1) {
      float mo = __shfl_xor(m, off);
      float so = __shfl_xor(sg, off);
      float mn = fmaxf(m, mo);
      sg = sg * __expf(m - mn) + so * __expf(mo - mn);
      m  = mn;
      dl += __shfl_xor(dl, off);
      cl += __shfl_xor(cl, off);
    }
    mrun[v] = m; srun[v] = sg; dlab[v] = dl; cxlab[v] = cl;
  }

  // ---- phase 4: per-pixel NLL + entailment (2 lanes/wave own 8 rows each) -
  float accN = 0.0f, accE = 0.0f, accC = 0.0f;
  if (lm == 0) {
#pragma unroll
    for (int v = 0; v < 8; ++v) {
      int   pm  = wave * 16 + v + 8 * hi;
      float vld = vld_l[pm];
      float lse = mrun[v] + __logf(srun[v]);
      float nll = lse + LOGIT_SCALE * dlab[v];
      int   lab = labv[v];
      float pt  = yt_l[lab];
      float pnm = pn_l[lab];
      float apr = ap_l[lab];
      float cxa = -cxlab[v];                       // Lorentz inner for angle
      float num = fmaf(cxa, pt, xtv[v]);
      float den = pnm * sqrtf(fmaxf(fmaf(cxa, cxa, -1.0f), 0.0f));
      float ai  = num / (den + F_EPS);
      ai = fminf(fmaxf(ai, -1.0f + F_EPS), 1.0f - F_EPS);
      float ent = fmaxf(acosf(ai) - apr, 0.0f);
      accN = fmaf(nll, vld, accN);
      accE = fmaf(ent, vld, accE);
      accC += vld;
    }
  }

  red[tid * 3 + 0] = accN;
  red[tid * 3 + 1] = accE;
  red[tid * 3 + 2] = accC;
  __syncthreads();
  for (int st = 128; st > 0; st >>= 1) {
    if (tid < st) {
      red[tid * 3 + 0] += red[(tid + st) * 3 + 0];
      red[tid * 3 + 1] += red[(tid + st) * 3 + 1];
      red[tid * 3 + 2] += red[(tid + st) * 3 + 2];
    }
    __syncthreads();
  }
  if (tid == 0) {
    partials[blockIdx.x * 3 + 0] = red[0];
    partials[blockIdx.x * 3 + 1] = red[1];
    partials[blockIdx.x * 3 + 2] = red[2];
  }
}

// ---------------------------------------------------------------------------
// Kernel 3: fixed-order reduction of per-WG partials -> scalar loss.
// ---------------------------------------------------------------------------
__global__ void meru_final(const float* __restrict__ partials, int nwg,
                           float* __restrict__ out) {
  __shared__ float red[256 * 3];
  int tid = threadIdx.x;
  float a = 0.0f, b = 0.0f, c = 0.0f;
  for (int i = tid; i < nwg; i += 256) {
    a += partials[i * 3 + 0];
    b += partials[i * 3 + 1];
    c += partials[i * 3 + 2];
  }
  red[tid * 3 + 0] = a; red[tid * 3 + 1] = b; red[tid * 3 + 2] = c;
  __syncthreads();
  for (int st = 128; st > 0; st >>= 1) {
    if (tid < st) {
      red[tid * 3 + 0] += red[(tid + st) * 3 + 0];
      red[tid * 3 + 1] += red[(tid + st) * 3 + 1];
      red[tid * 3 + 2] += red[(tid + st) * 3 + 2];
    }
    __syncthreads();
  }
  if (tid == 0) {
    float nv = fmaxf(red[2], 1.0f);
    out[0] = red[0] / nv + ENT_W * (red[1] / nv);
  }
}

// ---------------------------------------------------------------------------
extern "C" void kernel_launch(void* const* d_in, const int* in_sizes, int n_in,
                              void* d_out, int out_size, void* d_ws,
                              size_t ws_size, hipStream_t stream) {
  (void)n_in; (void)out_size; (void)ws_size;
  const float*         feats  = (const float*)d_in[0];
  const float*         tp     = (const float*)d_in[1];
  const int*           labels = (const int*)d_in[2];
  const unsigned char* maskp  = (const unsigned char*)d_in[3];  // jnp bool = 1B
  float* out = (float*)d_out;
  float* ws  = (float*)d_ws;

  // ws layout (floats): [0,10240) txt | [10240,10400) y_time |
  // [10400,10560) |txt| | [10560,10720) aperture | [10720,...) partials
  float* partials = ws + C_PAD * D_DIM + 3 * C_PAD;

  const int npix = in_sizes[2];            // B*H*W = 524288
  const int nwg  = npix / PIX_PER_WG;      // 4096

  meru_protos<<<dim3(1), dim3(C_PAD), 0, stream>>>(tp, ws);
  meru_main<<<dim3(nwg), dim3(256), 0, stream>>>(feats, labels, maskp, ws,
                                                 partials);
  meru_final<<<dim3(1), dim3(256), 0, stream>>>(partials, nwg, out);
}